// multiHeadAttention_61091614819027
// MI455X (gfx1250) — compile-verified
//
#include <hip/hip_runtime.h>
#include <hip/hip_bf16.h>

typedef float v2f __attribute__((ext_vector_type(2)));
typedef float v8f __attribute__((ext_vector_type(8)));

namespace {
constexpr int   kBatch   = 4;
constexpr int   kSeq     = 2048;
constexpr int   kHeads   = 8;
constexpr int   kHeadDim = 4;
constexpr int   kEmbed   = 32;
constexpr int   kRows    = kBatch * kSeq;          // 8192
constexpr float kScale   = 0.17677669529663687f;   // 1/sqrt(32)
constexpr float kMasked  = -1e20f * kScale;        // == (-1e20)/sqrt(32)
}

// ---------------------------------------------------------------------------
// y = x @ W + b for x:(rows,32), W:(32,32), b:(32,)
// One wave handles a 16-row tile: 2 column tiles x 8 K-chunks of 4
// => 16 x v_wmma_f32_16x16x4_f32 per wave.
// A-frag (16x4 f32): lane holds row M=lane&15, dims {d0,d0+1}, d0=2*(lane>>4)
// B-frag (4x16 f32): lane holds col N=lane&15, dims {d0,d0+1}
// C/D (16x16 f32):   lane holds col N=lane&15, VGPR j -> row M=j+8*(lane>>4)
// ---------------------------------------------------------------------------
__global__ __launch_bounds__(128) void linear32_wmma(
    const float* __restrict__ X, const float* __restrict__ W,
    const float* __restrict__ bias, float* __restrict__ Y, int rows)
{
  const int lane = threadIdx.x & 31;
  const int wid  = (blockIdx.x * blockDim.x + threadIdx.x) >> 5;
  const int r0   = wid * 16;
  if (r0 >= rows) return;                 // wave-uniform exit
  const int col  = lane & 15;
  const int half = lane >> 4;
  const int d0   = half * 2;

  #pragma unroll
  for (int c0 = 0; c0 < kEmbed; c0 += 16) {
    v8f acc = {};
    #pragma unroll
    for (int k0 = 0; k0 < kEmbed; k0 += 4) {
      v2f a = *(const v2f*)(X + (size_t)(r0 + col) * kEmbed + k0 + d0);
      v2f b;
      b.x = W[(size_t)(k0 + d0)     * kEmbed + c0 + col];
      b.y = W[(size_t)(k0 + d0 + 1) * kEmbed + c0 + col];
      acc = __builtin_amdgcn_wmma_f32_16x16x4_f32(false, a, false, b,
                                                  (short)0, acc, false, false);
    }
    const float bv = bias[c0 + col];
    #pragma unroll
    for (int j = 0; j < 8; ++j)
      Y[(size_t)(r0 + j + 8 * half) * kEmbed + c0 + col] = acc[j] + bv;
  }
}

// ---------------------------------------------------------------------------
// Flash attention: one wave per (n, h, 16-query tile). K streamed in chunks
// of 16 columns; S tile via one fp32 WMMA (K-dim == head_dim == 4).
// Q fragment is pre-scaled by 1/sqrt(32) so S arrives already scaled.
// Split-softmax: each lane owns columns k % 16 == lane&15 and keeps private
// (m, l, acc) with a single-exp online update; exact merge across the
// 16-lane group at the end.
// ---------------------------------------------------------------------------
__global__ __launch_bounds__(128) void attn_flash_wmma(
    const float* __restrict__ Qp, const float* __restrict__ Kp,
    const float* __restrict__ Vp, const int* __restrict__ mask,
    float* __restrict__ Out)
{
  const int lane = threadIdx.x & 31;
  const int wid  = (blockIdx.x * blockDim.x + threadIdx.x) >> 5;
  const int qt = wid & (kSeq / 16 - 1);        // 0..127
  const int h  = (wid >> 7) & (kHeads - 1);    // 0..7
  const int n  = wid >> 10;                    // 0..3
  const int q0 = qt * 16;

  const int col  = lane & 15;
  const int half = lane >> 4;
  const int d0   = half * 2;

  // A fragment: Q rows q0..q0+15, this lane's two head-dims, pre-scaled.
  v2f a = *(const v2f*)(Qp +
      ((size_t)(n * kSeq + q0 + col) * kHeads + h) * kHeadDim + d0);
  a.x *= kScale;
  a.y *= kScale;

  float m[8], lsum[8], acc[8][4];
  #pragma unroll
  for (int j = 0; j < 8; ++j) {
    m[j] = -INFINITY; lsum[j] = 0.f;
    acc[j][0] = acc[j][1] = acc[j][2] = acc[j][3] = 0.f;
  }

  const size_t kvRowBase = (size_t)n * kSeq;
  const size_t maskBase  = (size_t)n * kSeq * kSeq;

  #pragma unroll 2
  for (int k0 = 0; k0 < kSeq; k0 += 16) {
    const size_t kvrow = (kvRowBase + k0 + col) * kHeads + h;
    const v2f    b  = *(const v2f*)(Kp + kvrow * kHeadDim + d0);
    const float4 vv = *(const float4*)(Vp + kvrow * kHeadDim);
    if (k0 + 16 < kSeq) {   // prefetch next chunk (global_prefetch_b8)
      __builtin_prefetch(Kp + (kvrow + (size_t)16 * kHeads) * kHeadDim, 0, 3);
      __builtin_prefetch(Vp + (kvrow + (size_t)16 * kHeads) * kHeadDim, 0, 3);
    }

    v8f s8 = {};
    s8 = __builtin_amdgcn_wmma_f32_16x16x4_f32(false, a, false, b,
                                               (short)0, s8, false, false);
    #pragma unroll
    for (int j = 0; j < 8; ++j) {
      const int qrow = q0 + j + 8 * half;
      const int mk = mask[maskBase + (size_t)qrow * kSeq + k0 + col];
      const float s = (mk == 0) ? kMasked : s8[j];
      // Single-exp online softmax update:
      //   d>0 (new max): corr=e, p=1, m=s;  d<=0: corr=1, p=e
      // d=+inf when m=-inf -> e=0, p=1 (first sample); no NaN path.
      const float d = s - m[j];
      const float e = __expf(-fabsf(d));
      const bool  gt = d > 0.f;
      const float corr = gt ? e : 1.f;
      const float p    = gt ? 1.f : e;
      m[j] = gt ? s : m[j];
      lsum[j]   = lsum[j]   * corr + p;
      acc[j][0] = acc[j][0] * corr + p * vv.x;
      acc[j][1] = acc[j][1] * corr + p * vv.y;
      acc[j][2] = acc[j][2] * corr + p * vv.z;
      acc[j][3] = acc[j][3] * corr + p * vv.w;
    }
  }

  // Merge the 16 per-lane softmax splits (exact flash merge), then store.
  #pragma unroll
  for (int j = 0; j < 8; ++j) {
    float mt = m[j];
    mt = fmaxf(mt, __shfl_xor(mt, 1));
    mt = fmaxf(mt, __shfl_xor(mt, 2));
    mt = fmaxf(mt, __shfl_xor(mt, 4));
    mt = fmaxf(mt, __shfl_xor(mt, 8));
    const float corr = __expf(m[j] - mt);
    float lj = lsum[j] * corr;
    float o0 = acc[j][0] * corr, o1 = acc[j][1] * corr;
    float o2 = acc[j][2] * corr, o3 = acc[j][3] * corr;
    #pragma unroll
    for (int off = 1; off < 16; off <<= 1) {
      lj += __shfl_xor(lj, off);
      o0 += __shfl_xor(o0, off);
      o1 += __shfl_xor(o1, off);
      o2 += __shfl_xor(o2, off);
      o3 += __shfl_xor(o3, off);
    }
    if (col == 0) {
      const float inv = 1.0f / lj;
      float4 o = make_float4(o0 * inv, o1 * inv, o2 * inv, o3 * inv);
      *(float4*)(Out +
          ((size_t)(n * kSeq + q0 + j + 8 * half) * kHeads + h) * kHeadDim) = o;
    }
  }
}

// ---------------------------------------------------------------------------
extern "C" void kernel_launch(void* const* d_in, const int* in_sizes, int n_in,
                              void* d_out, int out_size, void* d_ws, size_t ws_size,
                              hipStream_t stream) {
  (void)in_sizes; (void)n_in; (void)out_size; (void)ws_size;
  const float* query = (const float*)d_in[0];
  const float* key   = (const float*)d_in[1];
  const float* value = (const float*)d_in[2];
  const int*   mask  = (const int*)  d_in[3];
  const float* Wq = (const float*)d_in[4];  const float* bq = (const float*)d_in[5];
  const float* Wk = (const float*)d_in[6];  const float* bk = (const float*)d_in[7];
  const float* Wv = (const float*)d_in[8];  const float* bv = (const float*)d_in[9];
  const float* Wo = (const float*)d_in[10]; const float* bo = (const float*)d_in[11];
  float* out = (float*)d_out;

  const size_t mat = (size_t)kRows * kEmbed;   // 262144 floats = 1 MB
  float* Qp = (float*)d_ws;
  float* Kp = Qp + mat;
  float* Vp = Kp + mat;
  float* Ao = Vp + mat;

  const int linBlocks  = (kRows / 16) / 4;                              // 128
  const int attnBlocks = (kBatch * kHeads * (kSeq / 16)) / 4;           // 1024

  linear32_wmma<<<linBlocks, 128, 0, stream>>>(query, Wq, bq, Qp, kRows);
  linear32_wmma<<<linBlocks, 128, 0, stream>>>(key,   Wk, bk, Kp, kRows);
  linear32_wmma<<<linBlocks, 128, 0, stream>>>(value, Wv, bv, Vp, kRows);
  attn_flash_wmma<<<attnBlocks, 128, 0, stream>>>(Qp, Kp, Vp, mask, Ao);
  linear32_wmma<<<linBlocks, 128, 0, stream>>>(Ao, Wo, bo, out, kRows);
}